// Experts_31121333027220
// MI455X (gfx1250) — compile-verified
//
#include <hip/hip_runtime.h>
#include <hip/hip_bf16.h>

// ---------------------------------------------------------------------------
// CDNA5 (gfx1250) implementation of the K-expert causal-GNN reference.
// All GEMMs use V_WMMA_F32_16X16X4_F32 (full fp32 parity with reference).
// Hard-concrete masks in eval forward are exactly (logit > 0) ? 1 : 0.
// Edge segment-sums use a one-time CSR build; all 27 aggregation passes are
// atomic-free register-accumulated gathers.
// ---------------------------------------------------------------------------

typedef float v2f __attribute__((ext_vector_type(2)));
typedef float v8f __attribute__((ext_vector_type(8)));

__device__ __forceinline__ void atomAdd(float* p, float v) {
    __hip_atomic_fetch_add(p, v, __ATOMIC_RELAXED, __HIP_MEMORY_SCOPE_AGENT);
}

// ---------------------------------------------------------------------------
// fp32 WMMA GEMM:  OUT[m, c] = act(A'[m, :K] @ W[:K, 128] + bias[c])
//   MODE 0: A' = A          MODE 1: A' = (1 + eps) * A + Agg   (GIN pre)
//   ACT  0: ReLU            ACT  1: step (x > 0 ? 1 : 0)
//   RS: compile-time output row stride (constant store offsets).
// Block: 256 threads = 8 waves; 32 rows per block; wave w owns cols
// [16w,16w+16) for TWO 16-row tiles, so each B fragment feeds 2 WMMAs.
// ---------------------------------------------------------------------------
template<int MODE, int ACT, int K, int RS>
__launch_bounds__(256)
__global__ void gemm32x128(const float* __restrict__ A,
                           const float* __restrict__ Agg,
                           const float* __restrict__ epsP,
                           const float* __restrict__ W,     // [K][128] row-major
                           const float* __restrict__ bias,  // [128]
                           float* __restrict__ out, int M)
{
    static_assert(K % 4 == 0, "K must be multiple of 4");
    constexpr int PAD = 4;
    __shared__ float As[32][K + PAD];

    const int tid  = threadIdx.x;
    const int wave = tid >> 5;
    const int lane = tid & 31;
    const int row0 = blockIdx.x * 32;

    float epsv = 0.0f;
    if (MODE == 1) epsv = 1.0f + epsP[0];

    // Cooperative float4 A-tile staging (b128 loads + b128 LDS stores).
    constexpr int CH = 32 * (K / 4);
    for (int idx = tid; idx < CH; idx += 256) {
        int r  = idx / (K / 4);
        int k4 = (idx - r * (K / 4)) * 4;
        int m  = row0 + r;
        float4 v{0.f, 0.f, 0.f, 0.f};
        if (m < M) {
            v = *(const float4*)(A + (long)m * K + k4);
            if (MODE == 1) {
                const float4 g = *(const float4*)(Agg + (long)m * K + k4);
                v.x = epsv * v.x + g.x;
                v.y = epsv * v.y + g.y;
                v.z = epsv * v.z + g.z;
                v.w = epsv * v.w + g.w;
            }
        }
        *(float4*)(&As[r][k4]) = v;
    }
    __syncthreads();

    const int col0 = wave * 16;
    const int halo = lane >> 4;   // 0: K=k0,k0+1   1: K=k0+2,k0+3
    const int l16  = lane & 15;

    v8f acc0 = {}, acc1 = {};
#pragma unroll 4
    for (int k0 = 0; k0 < K; k0 += 4) {
        const int ka = k0 + 2 * halo;
        v2f a0, a1, b;
        a0.x = As[l16][ka];       a0.y = As[l16][ka + 1];
        a1.x = As[16 + l16][ka];  a1.y = As[16 + l16][ka + 1];
        b.x  = W[(long)ka * 128 + col0 + l16];
        b.y  = W[(long)(ka + 1) * 128 + col0 + l16];
        acc0 = __builtin_amdgcn_wmma_f32_16x16x4_f32(false, a0, false, b,
                                                     (short)0, acc0, false, false);
        acc1 = __builtin_amdgcn_wmma_f32_16x16x4_f32(false, a1, false, b,
                                                     (short)0, acc1, false, false);
    }

    const int   c  = col0 + l16;
    const float bv = bias[c];
    float* base = out + (long)(row0 + 8 * halo) * RS + c;
    auto actf = [](float v) {
        if (ACT == 0) return v > 0.0f ? v : 0.0f;
        return v > 0.0f ? 1.0f : 0.0f;
    };
    if (row0 + 32 <= M) {
#pragma unroll
        for (int r = 0; r < 8; ++r) {
            base[(long)r * RS]        = actf(acc0[r] + bv);
            base[(long)(16 + r) * RS] = actf(acc1[r] + bv);
        }
    } else {
#pragma unroll
        for (int r = 0; r < 8; ++r) {
            int m0 = row0 + r + 8 * halo;
            if (m0 < M)      base[(long)r * RS]        = actf(acc0[r] + bv);
            if (m0 + 16 < M) base[(long)(16 + r) * RS] = actf(acc1[r] + bv);
        }
    }
}

// ---------------------------------------------------------------------------
// Fused 2-layer expert MLP -> binary scalar mask (node / edge masks).
//   score[m] = step( relu(A'[m]@W1_k + b1_k) . w2_k + b2_k )
//   MODE 0: A' = A[m, :K]                        (node mask, K=128)
//   MODE 1: A' = concat(Z[src[m]], Z[dst[m]])    (edge mask, K=256)
// 32 rows per block (two tiles per wave, B reused); blockIdx.y = expert.
// Hidden activations never touch memory.
// ---------------------------------------------------------------------------
template<int MODE, int K>
__launch_bounds__(256)
__global__ void expert_score(const float* __restrict__ Zfeat,
                             const int* __restrict__ src,
                             const int* __restrict__ dst,
                             const float* __restrict__ W1b,  // [8][K][128]
                             const float* __restrict__ b1b,  // [8][128]
                             const float* __restrict__ W2b,  // [8][128]
                             const float* __restrict__ b2b,  // [8]
                             float* __restrict__ outB,       // index m*8 + kx
                             int M)
{
    constexpr int PAD = 4;
    __shared__ float As[32][K + PAD];
    __shared__ float psum[8][32];

    const int tid  = threadIdx.x;
    const int wave = tid >> 5;
    const int lane = tid & 31;
    const int row0 = blockIdx.x * 32;
    const int kx   = blockIdx.y;

    const float* W1 = W1b + (long)kx * K * 128;
    const float* b1 = b1b + (long)kx * 128;
    const float* w2 = W2b + (long)kx * 128;
    const float  b2 = b2b[kx];

    constexpr int CH = 32 * (K / 4);
    for (int idx = tid; idx < CH; idx += 256) {
        int r  = idx / (K / 4);
        int k4 = (idx - r * (K / 4)) * 4;
        int m  = row0 + r;
        float4 v{0.f, 0.f, 0.f, 0.f};
        if (m < M) {
            if (MODE == 0) {
                v = *(const float4*)(Zfeat + (long)m * K + k4);
            } else {
                int node = (k4 < 128) ? src[m] : dst[m];
                v = *(const float4*)(Zfeat + (long)node * 128 + (k4 & 127));
            }
        }
        *(float4*)(&As[r][k4]) = v;
    }
    __syncthreads();

    const int col0 = wave * 16;
    const int halo = lane >> 4;
    const int l16  = lane & 15;

    v8f acc0 = {}, acc1 = {};
#pragma unroll 4
    for (int k0 = 0; k0 < K; k0 += 4) {
        const int ka = k0 + 2 * halo;
        v2f a0, a1, b;
        a0.x = As[l16][ka];       a0.y = As[l16][ka + 1];
        a1.x = As[16 + l16][ka];  a1.y = As[16 + l16][ka + 1];
        b.x  = W1[(long)ka * 128 + col0 + l16];
        b.y  = W1[(long)(ka + 1) * 128 + col0 + l16];
        acc0 = __builtin_amdgcn_wmma_f32_16x16x4_f32(false, a0, false, b,
                                                     (short)0, acc0, false, false);
        acc1 = __builtin_amdgcn_wmma_f32_16x16x4_f32(false, a1, false, b,
                                                     (short)0, acc1, false, false);
    }

    const int   c   = col0 + l16;
    const float b1v = b1[c];
    const float w2v = w2[c];
#pragma unroll
    for (int r = 0; r < 8; ++r) {
        float h0 = acc0[r] + b1v; h0 = h0 > 0.f ? h0 : 0.f;
        float h1 = acc1[r] + b1v; h1 = h1 > 0.f ? h1 : 0.f;
        float p0 = h0 * w2v, p1 = h1 * w2v;
        // sum over the 16 lanes of this half-wave (xor<=8 stays in the half)
        p0 += __shfl_xor(p0, 1, 32); p1 += __shfl_xor(p1, 1, 32);
        p0 += __shfl_xor(p0, 2, 32); p1 += __shfl_xor(p1, 2, 32);
        p0 += __shfl_xor(p0, 4, 32); p1 += __shfl_xor(p1, 4, 32);
        p0 += __shfl_xor(p0, 8, 32); p1 += __shfl_xor(p1, 8, 32);
        if (l16 == 0) {
            psum[wave][r + 8 * halo]      = p0;
            psum[wave][16 + r + 8 * halo] = p1;
        }
    }
    __syncthreads();

    if (tid < 32) {
        float s = b2;
#pragma unroll
        for (int w = 0; w < 8; ++w) s += psum[w][tid];
        int m = row0 + tid;
        if (m < M) outB[(long)m * 8 + kx] = s > 0.0f ? 1.0f : 0.0f;
    }
}

// ---------------------------------------------------------------------------
// One-time CSR build (dst-sorted edge index) for atomic-free aggregation.
// ---------------------------------------------------------------------------
__global__ void hist_dst(const int* __restrict__ dst, int* __restrict__ cnt, int E)
{
    int e = blockIdx.x * blockDim.x + threadIdx.x;
    if (e < E) atomicAdd(&cnt[dst[e]], 1);
}

// Single-block inclusive-scan -> exclusive rowptr + cursor copy.
__global__ void scan_rowptr(const int* __restrict__ cnt, int* __restrict__ rowptr,
                            int* __restrict__ cursor, int N)
{
    __shared__ int buf[1024];
    const int tid = threadIdx.x;
    int carry = 0;
    if (tid == 0) rowptr[0] = 0;
    for (int base = 0; base < N; base += 1024) {
        int i = base + tid;
        int v = (i < N) ? cnt[i] : 0;
        buf[tid] = v;
        __syncthreads();
        for (int off = 1; off < 1024; off <<= 1) {
            int t = (tid >= off) ? buf[tid - off] : 0;
            __syncthreads();
            buf[tid] += t;
            __syncthreads();
        }
        int total = buf[tid] + carry;
        if (i < N) { rowptr[i + 1] = total; cursor[i] = total - v; }
        carry += buf[1023];
        __syncthreads();
    }
}

__global__ void csr_fill(const int* __restrict__ src, const int* __restrict__ dst,
                         int* __restrict__ cursor, int* __restrict__ csrc,
                         int* __restrict__ ceid, int E)
{
    int e = blockIdx.x * blockDim.x + threadIdx.x;
    if (e >= E) return;
    int p = atomicAdd(&cursor[dst[e]], 1);
    csrc[p] = src[e];
    ceid[p] = e;
}

// agg[n, :] = sum_{e: dst[e]==n} h[src[e], :] * ew[e]   (ew binary or null)
// Thread = (node, 4-feature chunk); register accumulation, single b128 store.
__global__ void agg_csr(const float* __restrict__ h,
                        const int* __restrict__ rowptr,
                        const int* __restrict__ csrc,
                        const int* __restrict__ ceid,
                        const float* __restrict__ ew,   // stride 8, may be null
                        float* __restrict__ agg, int Nn)
{
    long gid = (long)blockIdx.x * blockDim.x + threadIdx.x;
    if (gid >= (long)Nn * 32) return;
    int n = (int)(gid >> 5);
    int f = (int)(gid & 31) * 4;
    const int j0 = rowptr[n], j1 = rowptr[n + 1];
    float4 s{0.f, 0.f, 0.f, 0.f};
    for (int j = j0; j < j1; ++j) {
        if (ew && ew[(long)ceid[j] * 8] == 0.0f) continue;
        const float4 v = *(const float4*)(h + (long)csrc[j] * 128 + f);
        s.x += v.x; s.y += v.y; s.z += v.z; s.w += v.w;
    }
    *(float4*)(agg + (long)n * 128 + f) = s;
}

// masked_x[n,f] = x[n,f] * node_m[n] * feat_m[n,f]   (masks binary, from d_out)
__global__ void mask_x(const float* __restrict__ x,
                       const float* __restrict__ nodeM,  // pre-offset by k, stride 8
                       const float* __restrict__ featM,  // pre-offset by k*128, row stride 1024
                       float* __restrict__ out, int N)
{
    long idx = (long)blockIdx.x * blockDim.x + threadIdx.x;
    if (idx >= (long)N * 128) return;
    int n = (int)(idx >> 7), f = (int)(idx & 127);
    out[idx] = x[idx] * nodeM[(long)n * 8] * featM[(long)n * 1024 + f];
}

__global__ void count_nodes(const int* __restrict__ batch, float* __restrict__ counts, int N)
{
    int n = blockIdx.x * blockDim.x + threadIdx.x;
    if (n < N) atomAdd(&counts[batch[n]], 1.0f);
}

// out[batch[n]*rowStride + f] += h[n,f]   (out pre-offset by column base)
__global__ void pool_accum(const float* __restrict__ h,
                           const int* __restrict__ batch,
                           float* __restrict__ out, int rowStride, int N)
{
    long idx = (long)blockIdx.x * blockDim.x + threadIdx.x;
    if (idx >= (long)N * 128) return;
    int n = (int)(idx >> 7), f = (int)(idx & 127);
    atomAdd(out + (long)batch[n] * rowStride + f, h[idx]);
}

__global__ void scale_pool(float* __restrict__ p, const float* __restrict__ counts,
                           int perB, long total)
{
    long idx = (long)blockIdx.x * blockDim.x + threadIdx.x;
    if (idx >= total) return;
    float c = counts[idx / perB];
    c = c < 1.0f ? 1.0f : c;
    p[idx] /= c;
}

// expert_logits[b,k,c] = h_stable[b,k,:] @ clf_W[k,:,c] + clf_b[k,c]
__global__ void clf_head(const float* __restrict__ hs,    // [B,8,128]
                         const float* __restrict__ clfW,  // [8,128,10]
                         const float* __restrict__ clfb,  // [8,10]
                         float* __restrict__ out, int total)
{
    int idx = blockIdx.x * blockDim.x + threadIdx.x;
    if (idx >= total) return;
    int c = idx % 10, k = (idx / 10) % 8, b = idx / 80;
    const float* hv = hs + (long)b * 1024 + (long)k * 128;
    const float* wv = clfW + (long)k * 1280 + c;
    float s = clfb[k * 10 + c];
    for (int h = 0; h < 128; ++h) s += hv[h] * wv[(long)h * 10];
    out[idx] = s;
}

// ---------------------------------------------------------------------------
// Host orchestration.
// Input flattening (top-level dict insertion order; nested dicts sorted-key
// order W1, W2, b1, b2, eps):
//   0 x | 1 edge_index | 2 batch
//   3..17  enc_causal (3 layers x [W1,W2,b1,b2,eps])
//   18..32 enc_cls    (3 layers x [W1,W2,b1,b2,eps])
//   33..36 nm [W1,W2,b1,b2] | 37..40 em | 41..44 fm | 45 clf_W | 46 clf_b
// Output (flat f32): h_stable[B,8,128] | h_orig[B,128] | node_m[N,8] |
//   edge_m[E,8] | feat_m[N,8,128] | expert_logits[B,8,10]
// ---------------------------------------------------------------------------
extern "C" void kernel_launch(void* const* d_in, const int* in_sizes, int n_in,
                              void* d_out, int out_size, void* d_ws, size_t ws_size,
                              hipStream_t stream)
{
    const int N = 10000, E = 160000, H = 128, KE = 8, B = 128, C = 10;
    (void)in_sizes; (void)n_in; (void)out_size; (void)ws_size;

    const float* x     = (const float*)d_in[0];
    const int*   src   = (const int*)d_in[1];
    const int*   dst   = src + E;
    const int*   batch = (const int*)d_in[2];
    auto f32 = [&](int i) { return (const float*)d_in[i]; };

    // Workspace: Z + 4 N*H fp32 buffers + counts + CSR int buffers (~27 MB)
    float* Z   = (float*)d_ws;
    float* b0  = Z   + (long)N * H;
    float* bAg = b0  + (long)N * H;   // aggregation output
    float* bHd = bAg + (long)N * H;   // hidden buffer
    float* b3  = bHd + (long)N * H;
    float* counts = b3 + (long)N * H;       // [B] f32
    int* counts0  = (int*)(counts + B);     // [N]
    int* rowptr   = counts0 + N;            // [N+1]
    int* cursor   = rowptr + (N + 1);       // [N]
    int* csr_src  = cursor + N;             // [E]
    int* csr_eid  = csr_src + E;            // [E]

    float* out = (float*)d_out;
    const long OFF_HST = 0;                 // 128*8*128 = 131072
    const long OFF_HO  = 131072;            // 128*128   =  16384
    const long OFF_ND  = 147456;            // 10000*8   =  80000
    const long OFF_ED  = 227456;            // 160000*8  = 1280000
    const long OFF_FT  = 1507456;           // 10000*8*128 = 10240000
    const long OFF_LG  = 11747456;          // 128*8*10  =  10240

    const int gN32 = (N + 31) / 32;
    const int gE32 = (E + 31) / 32;
    const int gEb  = (E + 255) / 256;
    const int gAgg = (int)(((long)N * 32 + 255) / 256);
    const int gNF  = (int)(((long)N * 128 + 255) / 256);

    // Zero pooled-accumulator output regions, counts, CSR histogram.
    hipMemsetAsync(out, 0, (size_t)OFF_ND * sizeof(float), stream);
    hipMemsetAsync(counts, 0, B * sizeof(float), stream);
    hipMemsetAsync(counts0, 0, (size_t)N * sizeof(int), stream);
    count_nodes<<<(N + 255) / 256, 256, 0, stream>>>(batch, counts, N);

    // ---- one-time CSR build (reused by all 27 aggregation passes) ----------
    hist_dst<<<gEb, 256, 0, stream>>>(dst, counts0, E);
    scan_rowptr<<<1, 1024, 0, stream>>>(counts0, rowptr, cursor, N);
    csr_fill<<<gEb, 256, 0, stream>>>(src, dst, cursor, csr_src, csr_eid, E);

    // ---- enc_causal: Z = GIN(x) --------------------------------------------
    {
        const float* h = x;
        float* louts[3] = { b0, b3, Z };
        for (int l = 0; l < 3; ++l) {
            const float* W1  = f32(3 + 5 * l + 0);
            const float* W2  = f32(3 + 5 * l + 1);
            const float* bb1 = f32(3 + 5 * l + 2);
            const float* bb2 = f32(3 + 5 * l + 3);
            const float* eps = f32(3 + 5 * l + 4);
            agg_csr<<<gAgg, 256, 0, stream>>>(h, rowptr, csr_src, csr_eid, nullptr, bAg, N);
            gemm32x128<1, 0, 128, 128><<<gN32, 256, 0, stream>>>(h, bAg, eps, W1, bb1, bHd, N);
            gemm32x128<0, 0, 128, 128><<<gN32, 256, 0, stream>>>(bHd, nullptr, nullptr, W2, bb2, louts[l], N);
            h = louts[l];
        }
    }

    // ---- masks (binary, written straight into d_out) -----------------------
    expert_score<0, 128><<<dim3(gN32, KE), 256, 0, stream>>>(
        Z, nullptr, nullptr, f32(33), f32(35), f32(34), f32(36), out + OFF_ND, N);
    expert_score<1, 256><<<dim3(gE32, KE), 256, 0, stream>>>(
        Z, src, dst, f32(37), f32(39), f32(38), f32(40), out + OFF_ED, E);
    for (int k = 0; k < KE; ++k) {
        gemm32x128<0, 0, 128, 128><<<gN32, 256, 0, stream>>>(
            Z, nullptr, nullptr, f32(41) + (long)k * 128 * 128, f32(43) + (long)k * 128,
            bHd, N);
        gemm32x128<0, 1, 128, 1024><<<gN32, 256, 0, stream>>>(
            bHd, nullptr, nullptr, f32(42) + (long)k * 128 * 128, f32(44) + (long)k * 128,
            out + OFF_FT + (long)k * 128, N);
    }

    // ---- enc_cls per expert on masked inputs, then pool --------------------
    for (int k = 0; k < KE; ++k) {
        mask_x<<<gNF, 256, 0, stream>>>(x, out + OFF_ND + k, out + OFF_FT + (long)k * 128, b0, N);
        const float* h = b0;
        float* louts[3] = { b3, b0, b3 };
        for (int l = 0; l < 3; ++l) {
            const float* W1  = f32(18 + 5 * l + 0);
            const float* W2  = f32(18 + 5 * l + 1);
            const float* bb1 = f32(18 + 5 * l + 2);
            const float* bb2 = f32(18 + 5 * l + 3);
            const float* eps = f32(18 + 5 * l + 4);
            agg_csr<<<gAgg, 256, 0, stream>>>(h, rowptr, csr_src, csr_eid, out + OFF_ED + k, bAg, N);
            gemm32x128<1, 0, 128, 128><<<gN32, 256, 0, stream>>>(h, bAg, eps, W1, bb1, bHd, N);
            gemm32x128<0, 0, 128, 128><<<gN32, 256, 0, stream>>>(bHd, nullptr, nullptr, W2, bb2, louts[l], N);
            h = louts[l];
        }
        pool_accum<<<gNF, 256, 0, stream>>>(b3, batch, out + OFF_HST + (long)k * 128, 1024, N);
    }

    // ---- h_orig pool, mean scaling, classifier head ------------------------
    pool_accum<<<gNF, 256, 0, stream>>>(Z, batch, out + OFF_HO, 128, N);
    scale_pool<<<(B * KE * H + 255) / 256, 256, 0, stream>>>(out + OFF_HST, counts, KE * H, (long)B * KE * H);
    scale_pool<<<(B * H + 255) / 256, 256, 0, stream>>>(out + OFF_HO, counts, H, (long)B * H);
    clf_head<<<(B * KE * C + 255) / 256, 256, 0, stream>>>(out + OFF_HST, f32(45), f32(46), out + OFF_LG, B * KE * C);
}